// MeshVerticalLayer_56160992363152
// MI455X (gfx1250) — compile-verified
//
#include <hip/hip_runtime.h>
#include <cstdint>

#define MESH_N 1024   // columns per row
#define ROWS   8      // batch rows per block
#define TPB    256    // 8 wave32 waves

// ---------------------------------------------------------------------------
// CDNA5 async global->LDS copy (GLOBAL_LOAD_ASYNC_TO_LDS_B128, ASYNCcnt path)
// VDST = per-lane LDS byte address, VADDR = per-lane 64-bit global address.
// Generic pointers to __shared__ have the LDS byte offset in their low 32 bits
// (ISA 10.2).
// ---------------------------------------------------------------------------
__device__ __forceinline__ void async_copy_16B(const float* gsrc, const float* ldst) {
    unsigned laddr = (unsigned)(uintptr_t)ldst;
    asm volatile("global_load_async_to_lds_b128 %0, %1, off"
                 :: "v"(laddr), "v"(gsrc)
                 : "memory");
}

__device__ __forceinline__ void wait_async_zero() {
    asm volatile("s_wait_asynccnt 0" ::: "memory");
}

// ---------------------------------------------------------------------------
// Setup: fuse right-perm, pairwise-perm, left-perm and the two coefficient
// tables into batch-invariant per-column data (runs once per launch, 1024 wi).
//   m = right[n]; p = pair[m]
//   idx1[n] = left[m]; idx2[n] = left[p]
//   coef[n] = (diag_re[m], diag_im[m], off_re[p], off_im[p])
// ---------------------------------------------------------------------------
__global__ void mesh_vlayer_setup(const float* __restrict__ diag,
                                  const float* __restrict__ offd,
                                  const int*   __restrict__ pairw,
                                  const int*   __restrict__ leftp,
                                  const int*   __restrict__ rightp,
                                  float4*      __restrict__ coef,
                                  int*         __restrict__ idx1,
                                  int*         __restrict__ idx2) {
    int n = blockIdx.x * blockDim.x + threadIdx.x;
    if (n < MESH_N) {
        int m = rightp[n];
        int p = pairw[m];
        idx1[n] = leftp[m];
        idx2[n] = leftp[p];
        coef[n] = make_float4(diag[m], diag[MESH_N + m], offd[p], offd[MESH_N + p]);
    }
}

// ---------------------------------------------------------------------------
// Main: each block streams ROWS batch rows. Row planes (re,im: 8KB) are
// double-buffered in LDS via async b128 copies; compute gathers from LDS.
// ---------------------------------------------------------------------------
__global__ __launch_bounds__(TPB) void mesh_vlayer_main(
        const float*  __restrict__ x,
        const float4* __restrict__ coef,
        const int4*   __restrict__ idx1v,
        const int4*   __restrict__ idx2v,
        float*        __restrict__ out,
        int nbatch) {
    __shared__ float lds[2][2 * MESH_N];   // [buf][ re(0..N-1) | im(N..2N-1) ]

    const int t = threadIdx.x;
    const long long rowBase = (long long)blockIdx.x * ROWS;
    const long long plane   = (long long)nbatch * MESH_N;
    const float* xre = x;
    const float* xim = x + plane;
    float* ore = out;
    float* oim = out + plane;

    // per-thread columns n in [4t, 4t+4): indices + coefficients in registers
    const int4   i1 = idx1v[t];
    const int4   i2 = idx2v[t];
    const float4 c0 = coef[4 * t + 0];
    const float4 c1 = coef[4 * t + 1];
    const float4 c2 = coef[4 * t + 2];
    const float4 c3 = coef[4 * t + 3];

    // prefetch row 0 into buffer 0
    {
        const long long r0 = rowBase;
        async_copy_16B(xre + r0 * MESH_N + 4 * t, &lds[0][4 * t]);
        async_copy_16B(xim + r0 * MESH_N + 4 * t, &lds[0][MESH_N + 4 * t]);
    }

    for (int r = 0; r < ROWS; ++r) {
        wait_async_zero();   // own wave's async copies for row r landed in LDS
        __syncthreads();     // everyone's landed; everyone done reading buf r-1

        if (r + 1 < ROWS) {  // safe post-barrier: nobody still reads buf (r+1)&1
            const long long rn = rowBase + r + 1;
            const int buf = (r + 1) & 1;
            async_copy_16B(xre + rn * MESH_N + 4 * t, &lds[buf][4 * t]);
            async_copy_16B(xim + rn * MESH_N + 4 * t, &lds[buf][MESH_N + 4 * t]);
        }

        const float* lre = &lds[r & 1][0];
        const float* lim = &lds[r & 1][MESH_N];

        float4 vre, vim;
        {
            float ar = lre[i1.x], ai = lim[i1.x], br = lre[i2.x], bi = lim[i2.x];
            vre.x = ar * c0.x - ai * c0.y + br * c0.z - bi * c0.w;
            vim.x = ar * c0.y + ai * c0.x + br * c0.w + bi * c0.z;
        }
        {
            float ar = lre[i1.y], ai = lim[i1.y], br = lre[i2.y], bi = lim[i2.y];
            vre.y = ar * c1.x - ai * c1.y + br * c1.z - bi * c1.w;
            vim.y = ar * c1.y + ai * c1.x + br * c1.w + bi * c1.z;
        }
        {
            float ar = lre[i1.z], ai = lim[i1.z], br = lre[i2.z], bi = lim[i2.z];
            vre.z = ar * c2.x - ai * c2.y + br * c2.z - bi * c2.w;
            vim.z = ar * c2.y + ai * c2.x + br * c2.w + bi * c2.z;
        }
        {
            float ar = lre[i1.w], ai = lim[i1.w], br = lre[i2.w], bi = lim[i2.w];
            vre.w = ar * c3.x - ai * c3.y + br * c3.z - bi * c3.w;
            vim.w = ar * c3.y + ai * c3.x + br * c3.w + bi * c3.z;
        }

        const long long row = rowBase + r;
        *(float4*)(ore + row * MESH_N + 4 * t) = vre;   // global_store_b128
        *(float4*)(oim + row * MESH_N + 4 * t) = vim;   // global_store_b128
    }
}

// ---------------------------------------------------------------------------
// Harness entry point.
// Inputs: x[2,B,N] f32, diag[2,N] f32, off_diag[2,N] f32,
//         pairwise_perm_idx[N] i32, left_perm_idx[N] i32, right_perm_idx[N] i32
// Output: [2,B,N] f32
// Workspace layout: coef float4[N] (16KB) | idx1 int[N] (4KB) | idx2 int[N] (4KB)
// ---------------------------------------------------------------------------
extern "C" void kernel_launch(void* const* d_in, const int* in_sizes, int n_in,
                              void* d_out, int out_size, void* d_ws, size_t ws_size,
                              hipStream_t stream) {
    (void)n_in; (void)out_size; (void)ws_size;

    const float* x     = (const float*)d_in[0];
    const float* diag  = (const float*)d_in[1];
    const float* offd  = (const float*)d_in[2];
    const int*   pairw = (const int*)d_in[3];
    const int*   leftp = (const int*)d_in[4];
    const int*   rightp= (const int*)d_in[5];

    float4* coef = (float4*)d_ws;
    int*    idx1 = (int*)((char*)d_ws + sizeof(float4) * MESH_N);
    int*    idx2 = idx1 + MESH_N;

    const int nbatch = in_sizes[0] / (2 * MESH_N);   // 32768

    mesh_vlayer_setup<<<MESH_N / TPB, TPB, 0, stream>>>(
        diag, offd, pairw, leftp, rightp, coef, idx1, idx2);

    mesh_vlayer_main<<<nbatch / ROWS, TPB, 0, stream>>>(
        x, coef, (const int4*)idx1, (const int4*)idx2, (float*)d_out, nbatch);
}